// ChildSumTreeLSTMOp_25323127177890
// MI455X (gfx1250) — compile-verified
//
#include <hip/hip_runtime.h>

#define HS 64
#define XS 128
#define NNODE 100000
#define NEDGE 1600000

typedef __attribute__((ext_vector_type(2))) float v2f;
typedef __attribute__((ext_vector_type(8))) float v8f;

__device__ __forceinline__ float sigmoidf_(float v) {
  return 1.0f / (1.0f + __expf(-v));
}
__device__ __forceinline__ float tanhf_(float v) {
  v = fminf(fmaxf(v, -15.0f), 15.0f);
  float e = __expf(2.0f * v);
  return (e - 1.0f) / (e + 1.0f);
}

// One K=4 step of fp32 WMMA: A frag = contiguous float2 of row (m0+lr),
// k = k0 + 2*half; B frag = contiguous float2 of W row (n0+lr) (B = W^T).
__device__ __forceinline__ v8f wmma_f32_step(const float* aPtr, const float* bPtr, v8f acc) {
  v2f a = *reinterpret_cast<const v2f*>(aPtr);
  v2f b = *reinterpret_cast<const v2f*>(bPtr);
  return __builtin_amdgcn_wmma_f32_16x16x4_f32(
      /*neg_a=*/false, a, /*neg_b=*/false, b,
      /*c_mod=*/(short)0, acc, /*reuse_a=*/false, /*reuse_b=*/false);
}

// ---------------- Kernel 1: iouf = x @ W_iouf^T  (N x 256, K = 128) ----------------
// Block = 256 threads (8 waves). Block handles 16 rows; wave w -> cols [16w,16w+16).
__global__ __launch_bounds__(256) void k1_node_gemm(
    const float* __restrict__ x, const float* __restrict__ W,
    float* __restrict__ iouf) {
  const int tid  = threadIdx.x;
  const int wave = tid >> 5, lane = tid & 31;
  const int lr   = lane & 15, half = lane >> 4;
  const int m0   = blockIdx.x * 16;
  const int n0   = wave * 16;

  const float* aP = x + (m0 + lr) * XS + 2 * half;
  const float* bP = W + (n0 + lr) * XS + 2 * half;

  v8f acc = {};
#pragma unroll
  for (int s = 0; s < XS / 4; ++s)
    acc = wmma_f32_step(aP + 4 * s, bP + 4 * s, acc);

#pragma unroll
  for (int d = 0; d < 8; ++d)
    iouf[(m0 + d + half * 8) * 256 + n0 + lr] = acc[d];
}

// ---------------- Kernel 2: per-edge f-gate GEMM + both segment sums ----------------
// f_lin = h_child[src] @ U_f_w^T (E x 64, K = 64)
// f_e = sigmoid(f_lin + f_x[dst] + U_f_b); c_agg[dst] += f_e * c_child[src];
// h_tild[dst] += h_child[src].
// Block = 256 threads (8 waves) = 32 edges: 2 groups of 16 edges x 4 col-tiles.
__global__ __launch_bounds__(256) void k2_edge(
    const float* __restrict__ h_child, const float* __restrict__ c_child,
    const float* __restrict__ iouf, const float* __restrict__ Ufw,
    const float* __restrict__ Ufb, const int* __restrict__ src,
    const int* __restrict__ dst, float* __restrict__ h_tild,
    float* __restrict__ c_agg) {
  const int tid  = threadIdx.x;
  const int wave = tid >> 5, lane = tid & 31;
  const int lr   = lane & 15, half = lane >> 4;
  const int g    = wave >> 2;   // edge group 0/1
  const int t    = wave & 3;    // col tile 0..3
  const int e0   = blockIdx.x * 32 + g * 16;

  const int   sRow = src[e0 + lr];
  const float* aP  = h_child + sRow * HS + 2 * half;          // gathered A rows
  const float* bP  = Ufw + (t * 16 + lr) * HS + 2 * half;     // B = U_f_w^T

  v8f acc = {};
#pragma unroll
  for (int s = 0; s < HS / 4; ++s)
    acc = wmma_f32_step(aP + 4 * s, bP + 4 * s, acc);

  const int   col = t * 16 + lr;
  const float fb  = Ufb[col];
#pragma unroll
  for (int d = 0; d < 8; ++d) {
    const int e  = e0 + d + half * 8;
    const int sr = src[e];
    const int dr = dst[e];
    const float fx = iouf[dr * 256 + 192 + col];   // f_x[dst]
    const float fe = sigmoidf_(acc[d] + fx + fb);
    const float cs = c_child[sr * HS + col];
    unsafeAtomicAdd(c_agg + dr * HS + col, fe * cs);
    unsafeAtomicAdd(h_tild + dr * HS + col, h_child[sr * HS + col]);
  }
}

// ---------------- Kernel 3: iou GEMM + nonlinearity epilogue ----------------
// iou = iou_x + h_tild @ U_iou^T + b_iou (N x 192, K = 64), then
// c = sig(i)*tanh(u) + c_agg; h = sig(o)*tanh(c); out = [h, c].
// Block = 384 threads (12 waves): 16 rows x 192 cols, one 16x16 tile per wave.
__global__ __launch_bounds__(384) void k3_final(
    const float* __restrict__ h_tild, const float* __restrict__ c_agg,
    const float* __restrict__ iouf, const float* __restrict__ Uiou,
    const float* __restrict__ biou, float* __restrict__ out) {
  __shared__ float sIou[16 * 192];

  const int tid  = threadIdx.x;
  const int wave = tid >> 5, lane = tid & 31;
  const int lr   = lane & 15, half = lane >> 4;
  const int m0   = blockIdx.x * 16;
  const int n0   = wave * 16;   // 0..176

  const float* aP = h_tild + (m0 + lr) * HS + 2 * half;
  const float* bP = Uiou + (n0 + lr) * HS + 2 * half;

  v8f acc = {};
#pragma unroll
  for (int s = 0; s < HS / 4; ++s)
    acc = wmma_f32_step(aP + 4 * s, bP + 4 * s, acc);

  const int   col = n0 + lr;
  const float bb  = biou[col];
#pragma unroll
  for (int d = 0; d < 8; ++d) {
    const int r = d + half * 8;
    sIou[r * 192 + col] = acc[d] + iouf[(m0 + r) * 256 + col] + bb;
  }
  __syncthreads();

  for (int idx = tid; idx < 16 * HS; idx += 384) {
    const int r = idx >> 6, c = idx & 63;
    const int row = m0 + r;
    const float iv = sIou[r * 192 + c];
    const float ov = sIou[r * 192 + 64 + c];
    const float uv = sIou[r * 192 + 128 + c];
    const float cv = sigmoidf_(iv) * tanhf_(uv) + c_agg[row * HS + c];
    const float hv = sigmoidf_(ov) * tanhf_(cv);
    out[row * 128 + c]      = hv;
    out[row * 128 + 64 + c] = cv;
  }
}

extern "C" void kernel_launch(void* const* d_in, const int* in_sizes, int n_in,
                              void* d_out, int out_size, void* d_ws, size_t ws_size,
                              hipStream_t stream) {
  const float* x       = (const float*)d_in[0];
  const float* h_child = (const float*)d_in[1];
  const float* c_child = (const float*)d_in[2];
  const float* W_iouf  = (const float*)d_in[3];
  const float* U_iou   = (const float*)d_in[4];
  const float* b_iou   = (const float*)d_in[5];
  const float* U_f_w   = (const float*)d_in[6];
  const float* U_f_b   = (const float*)d_in[7];
  const int*   src     = (const int*)d_in[8];
  const int*   dst     = (const int*)d_in[9];
  float* outF = (float*)d_out;

  float* ws     = (float*)d_ws;
  float* iouf   = ws;                                   // N*256 floats
  float* h_tild = ws + (size_t)NNODE * 256;             // N*64
  float* c_agg  = h_tild + (size_t)NNODE * HS;          // N*64

  // Zero the two segment-sum accumulators (graph-capture-safe async memset).
  hipMemsetAsync(h_tild, 0, 2 * (size_t)NNODE * HS * sizeof(float), stream);

  k1_node_gemm<<<NNODE / 16, 256, 0, stream>>>(x, W_iouf, iouf);
  k2_edge<<<NEDGE / 32, 256, 0, stream>>>(h_child, c_child, iouf, U_f_w, U_f_b,
                                          src, dst, h_tild, c_agg);
  k3_final<<<NNODE / 16, 384, 0, stream>>>(h_tild, c_agg, iouf, U_iou, b_iou, outF);
}